// ModelNew_80908593923002
// MI455X (gfx1250) — compile-verified
//
#include <hip/hip_runtime.h>

typedef __attribute__((ext_vector_type(2))) float v2f;
typedef __attribute__((ext_vector_type(8))) float v8f;

// Problem constants
#define CIN_  32
#define COUT_ 64
// weight layout (CIN, COUT, 5,5,5): idx = ci*8000 + co*125 + klin

__global__ __launch_bounds__(256) void convt_pool_sum_kernel(
    const float* __restrict__ x, const float* __restrict__ w,
    const float* __restrict__ bias, float* __restrict__ out)
{
    __shared__ float xs[125 * 34];   // [zlin 0..124][ci 0..31], padded stride 34 (8B-aligned ci-pairs)
    __shared__ float red[COUT_ * 2]; // per-co running max, per parity-half

    const int tid  = threadIdx.x;
    const int lane = tid & 31;
    const int wv   = tid >> 5;               // 8 waves
    const int bi   = blockIdx.x;             // (b,pd,ph,pw) flattened == output index
    const int b  = bi / 500;
    const int pd = (bi % 500) / 100;
    const int ph = (bi % 100) / 10;
    const int pw = bi % 10;

    // ---- Stage x tile (zero-padded halo) into LDS: id=3pd-1+zd, ih=3ph-1+zh, iw=3pw-1+zw ----
    for (int i = tid; i < CIN_ * 125; i += 256) {
        const int ci = i / 125, zl = i % 125;
        const int zd = zl / 25, zh = (zl / 5) % 5, zw = zl % 5;
        const int id = 3 * pd - 1 + zd;
        const int ih = 3 * ph - 1 + zh;
        const int iw = 3 * pw - 1 + zw;
        float v = 0.0f;
        if ((unsigned)id < 16u && (unsigned)ih < 32u && (unsigned)iw < 32u)
            v = x[(((size_t)(b * CIN_ + ci) * 16 + id) * 32 + ih) * 32 + iw];
        xs[zl * 34 + ci] = v;
    }
    __syncthreads();

    const int half   = lane >> 4;            // lane half selects K pair (A/B fragment layout)
    const int ln     = lane & 15;            // M (=co) / N (=pixel) index within tile
    const int cobase = (wv & 3) * 16;        // 4 co-tiles cover COUT=64
    const int phalf  = wv >> 2;              // parities [0..3] or [4..7]
    const int co     = cobase + ln;

    // Pixel decode: 27 real pooling positions (ad',ah',aw' in 0..2), slots 27..31 padded
    const int pix0 = ln;
    const int pix1 = (ln + 16 > 26) ? 26 : (ln + 16);
    const int ad0 = pix0 / 9, ah0 = (pix0 / 3) % 3, aw0 = pix0 % 3;
    const int ad1 = pix1 / 9, ah1 = (pix1 / 3) % 3, aw1 = pix1 % 3;

    const float NINF = -__builtin_huge_valf();
    v8f m0 = {NINF, NINF, NINF, NINF, NINF, NINF, NINF, NINF};
    v8f m1 = m0;

    for (int j = 0; j < 4; ++j) {
        const int p  = phalf * 4 + j;
        const int rd = (p >> 2) & 1, rh = (p >> 1) & 1, rw = p & 1;
        const int ndc = 3 - rd, nhc = 3 - rh, nwc = 3 - rw;   // taps per dim for this parity
        v8f c0 = {0, 0, 0, 0, 0, 0, 0, 0};
        v8f c1 = c0;
        for (int td = 0; td < ndc; ++td)
            for (int th = 0; th < nhc; ++th)
                for (int tw = 0; tw < nwc; ++tw) {
                    // kd = rd + 2*td ; input offset = 1 - td  (zd = ad' + 2 - td, in [0,5))
                    const int klin = (rd + 2 * td) * 25 + (rh + 2 * th) * 5 + (rw + 2 * tw);
                    const float* wrow = w + (size_t)co * 125 + klin;
                    const int zl0 = ((ad0 + 2 - td) * 5 + (ah0 + 2 - th)) * 5 + (aw0 + 2 - tw);
                    const int zl1 = ((ad1 + 2 - td) * 5 + (ah1 + 2 - th)) * 5 + (aw1 + 2 - tw);
#pragma unroll
                    for (int s = 0; s < 8; ++s) {             // ci = kk, kk+1 ; K step of 4
                        const int kk = s * 4 + 2 * half;
                        v2f a;                                 // A 16x4 f32 layout: lane=M, K pair by lane-half
                        a.x = wrow[(size_t)kk * 8000];
                        a.y = wrow[(size_t)(kk + 1) * 8000];
                        const v2f b0 = *(const v2f*)&xs[zl0 * 34 + kk];
                        const v2f b1 = *(const v2f*)&xs[zl1 * 34 + kk];
                        c0 = __builtin_amdgcn_wmma_f32_16x16x4_f32(
                                 false, a, false, b0, (short)0, c0, false, false);
                        c1 = __builtin_amdgcn_wmma_f32_16x16x4_f32(
                                 false, a, false, b1, (short)0, c1, false, false);
                    }
                }
#pragma unroll
        for (int r = 0; r < 8; ++r) {         // max across parity classes (elementwise over [co,pixel])
            m0[r] = fmaxf(m0[r], c0[r]);
            m1[r] = fmaxf(m1[r], c1[r]);
        }
    }

    // Mask padded pixel slots 27..31 of N-tile 1 (branchless, EXEC stays clean)
#pragma unroll
    for (int r = 0; r < 8; ++r)
        m1[r] = (ln >= 11) ? NINF : m1[r];

    // Per-co max over all pixel slots: cross-lane max within 16-lane halves
    float vr[8];
#pragma unroll
    for (int r = 0; r < 8; ++r) {
        float v = fmaxf(m0[r], m1[r]);
        for (int d = 1; d < 16; d <<= 1)
            v = fmaxf(v, __shfl_xor(v, d, 32));
        vr[r] = v;
    }
    if (ln == 0) {                            // lanes 0 and 16 write (co for each lane-half)
#pragma unroll
        for (int r = 0; r < 8; ++r)
            red[(cobase + 8 * half + r) * 2 + phalf] = vr[r];
    }
    __syncthreads();

    // Wave 0: combine parity halves, add bias, sum 64 channels, store one float
    if (tid < 32) {
        const int cA = tid, cB = tid + 32;
        float s = bias[cA] + fmaxf(red[cA * 2], red[cA * 2 + 1])
                + bias[cB] + fmaxf(red[cB * 2], red[cB * 2 + 1]);
        for (int d = 1; d < 32; d <<= 1)
            s += __shfl_xor(s, d, 32);
        if (tid == 0)
            out[bi] = s;
    }
}

extern "C" void kernel_launch(void* const* d_in, const int* in_sizes, int n_in,
                              void* d_out, int out_size, void* d_ws, size_t ws_size,
                              hipStream_t stream) {
    (void)in_sizes; (void)n_in; (void)d_ws; (void)ws_size; (void)out_size;
    const float* x    = (const float*)d_in[0];
    const float* w    = (const float*)d_in[1];
    const float* bias = (const float*)d_in[2];
    float* out        = (float*)d_out;

    // One workgroup per final output cell: 16*5*10*10 = 8000
    convt_pool_sum_kernel<<<dim3(8000), dim3(256), 0, stream>>>(x, w, bias, out);
}